// IterativeNN_16114717294766
// MI455X (gfx1250) — compile-verified
//
#include <hip/hip_runtime.h>
#include <stdint.h>

#define NVARS   2048
#define NODES   4096
#define NLAYERS 8
#define FANIN   4
#define BATCH   8192
#define ROWS    4                      // batch rows per workgroup
#define BLOCK   512                    // 16 wave32 waves
#define NPT     (NODES / BLOCK)        // nodes per thread per layer = 8
#define IDX_N   (NLAYERS * NODES * FANIN)   // 131072 indices

typedef __attribute__((ext_vector_type(4))) int v4i;

#if defined(__has_builtin)
#if __has_builtin(__builtin_amdgcn_global_load_async_to_lds_b128)
#define HAVE_ASYNC_LDS 1
#endif
#endif

struct __align__(32) IdxQ { long long v[4]; };

// ---- index fetch: int64 (direct) or packed uint16 (via d_ws) --------------
__device__ __forceinline__ void load_idx4(const long long* p, int idx[4]) {
    IdxQ q = *(const IdxQ*)p;                 // 2x global_load_b128, coalesced
    idx[0] = (int)q.v[0]; idx[1] = (int)q.v[1];
    idx[2] = (int)q.v[2]; idx[3] = (int)q.v[3];
}
__device__ __forceinline__ void load_idx4(const unsigned short* p, int idx[4]) {
    ushort4 u = *(const ushort4*)p;           // 1x global_load_b64, coalesced
    idx[0] = u.x; idx[1] = u.y; idx[2] = u.z; idx[3] = u.w;
}

__global__ void pack_idx(const long long* __restrict__ src,
                         unsigned short* __restrict__ dst) {
    int i = blockIdx.x * blockDim.x + threadIdx.x;
    if (i < IDX_N) dst[i] = (unsigned short)src[i];
}

template <typename IT>
__global__ __launch_bounds__(BLOCK) void pc_kernel(
    const float* __restrict__ x,
    const unsigned char* __restrict__ marg,
    const IT* __restrict__ cidx,
    float* __restrict__ out)
{
    // Ping-pong node tables, layout [node][row] so one b128 covers 4 rows.
    __shared__ __align__(16) float bufA[NODES * ROWS];   // 64 KB
    __shared__ __align__(16) float bufB[NODES * ROWS];   // 64 KB

    const int tid = threadIdx.x;
    const int b0  = blockIdx.x * ROWS;

    // ---- Stage the 4 contiguous x rows (32 KB) into bufB ------------------
    const float* xrow = x + (size_t)b0 * NVARS;          // 16B-aligned
    const int nChunks = (ROWS * NVARS) / 4;              // 16B chunks = 2048
#if HAVE_ASYNC_LDS
    for (int ch = tid; ch < nChunks; ch += BLOCK) {
        __builtin_amdgcn_global_load_async_to_lds_b128(
            (v4i*)(xrow + ch * 4),
            (v4i*)(bufB + ch * 4),
            0, 0);
    }
#if __has_builtin(__builtin_amdgcn_s_wait_asynccnt)
    __builtin_amdgcn_s_wait_asynccnt(0);
#else
    asm volatile("s_wait_asynccnt 0" ::: "memory");
#endif
#else
    for (int ch = tid; ch < nChunks; ch += BLOCK)
        ((float4*)bufB)[ch] = ((const float4*)xrow)[ch];
#endif
    __syncthreads();

    // ---- Build leaves into bufA: [c][r] = x, [NVARS+c][r] = 1-x, marg->1 --
    for (int c = tid; c < NVARS; c += BLOCK) {
        const bool m = (marg[c] != 0);       // one u8 load per column
#pragma unroll
        for (int r = 0; r < ROWS; ++r) {
            float v = bufB[r * NVARS + c];
            bufA[c * ROWS + r]           = m ? 1.0f : v;
            bufA[(NVARS + c) * ROWS + r] = m ? 1.0f : (1.0f - v);
        }
    }
    __syncthreads();

    // ---- Layer loop: alternating product / sum over FAN_IN=4 gathers ------
    float* src = bufA;
    float* dst = bufB;
#pragma unroll 1
    for (int l = 0; l < NLAYERS; ++l) {
        const IT* lip = cidx + ((size_t)l * NODES) * FANIN;
#pragma unroll
        for (int j = 0; j < NPT; ++j) {
            const int n = tid + j * BLOCK;
            const IT* ip = lip + (size_t)n * FANIN;
            if (j + 1 < NPT)
                __builtin_prefetch(ip + (size_t)BLOCK * FANIN, 0, 0);
            int idx[4];
            load_idx4(ip, idx);
            const float4* s4 = (const float4*)src;           // ds_load_b128 x4
            float4 a = s4[idx[0]], b = s4[idx[1]];
            float4 c = s4[idx[2]], d = s4[idx[3]];
            float4 o;
            if ((l & 1) == 0) {
                o.x = a.x * b.x * c.x * d.x;
                o.y = a.y * b.y * c.y * d.y;
                o.z = a.z * b.z * c.z * d.z;
                o.w = a.w * b.w * c.w * d.w;
            } else {
                o.x = a.x + b.x + c.x + d.x;
                o.y = a.y + b.y + c.y + d.y;
                o.z = a.z + b.z + c.z + d.z;
                o.w = a.w + b.w + c.w + d.w;
            }
            ((float4*)dst)[n] = o;                           // ds_store_b128
        }
        __syncthreads();
        float* t = src; src = dst; dst = t;
    }
    // After 8 layers (8 swaps) final values live in src (== bufA).

    // ---- Reduce 4096 nodes per row -> out[b0 + r] -------------------------
    const int r = tid & (ROWS - 1);        // 0..3
    const int t = tid >> 2;                // 0..127
    float s = 0.0f;
    for (int n = t; n < NODES; n += BLOCK / ROWS)
        s += src[n * ROWS + r];
    float* red = dst;                      // free buffer now
    red[t * ROWS + r] = s;
    __syncthreads();
    for (int off = (BLOCK / ROWS) >> 1; off >= 1; off >>= 1) {
        if (t < off)
            red[t * ROWS + r] += red[(t + off) * ROWS + r];
        __syncthreads();
    }
    if (t == 0)
        out[b0 + r] = red[r];
}

extern "C" void kernel_launch(void* const* d_in, const int* in_sizes, int n_in,
                              void* d_out, int out_size, void* d_ws, size_t ws_size,
                              hipStream_t stream) {
    (void)in_sizes; (void)n_in; (void)out_size;
    const float*         x    = (const float*)d_in[0];
    const unsigned char* marg = (const unsigned char*)d_in[1];   // jnp bool -> 1 byte
    const long long*     cidx = (const long long*)d_in[2];       // int64
    float*               out  = (float*)d_out;                   // (8192,1) f32

    const bool use_packed = (ws_size >= (size_t)IDX_N * sizeof(unsigned short));
    if (use_packed) {
        unsigned short* idx16 = (unsigned short*)d_ws;
        pack_idx<<<dim3(IDX_N / 256), dim3(256), 0, stream>>>(cidx, idx16);
        pc_kernel<unsigned short>
            <<<dim3(BATCH / ROWS), dim3(BLOCK), 0, stream>>>(x, marg, idx16, out);
    } else {
        pc_kernel<long long>
            <<<dim3(BATCH / ROWS), dim3(BLOCK), 0, stream>>>(x, marg, cidx, out);
    }
}